// MS_GetT_Voxel_53472342835718
// MI455X (gfx1250) — compile-verified
//
#include <hip/hip_runtime.h>

// Problem constants from the reference
#define T_DIM 4
#define B_DIM 8
#define H_DIM 480
#define W_DIM 640

struct __align__(16) BatchParam {
    float t0;     // t[first] per batch
    float denom;  // t[last](bumped) - t0, or 1.0 if zero
    float bump;   // 1.0 if cnt>=2 && t[last]==t[first], else 0
    int   last;   // index of last point in this batch (bump applies to it)
};

// ---------------------------------------------------------------------------
// Kernel 1: zero the output grid (also pre-warms it into the 192MB L2)
// ---------------------------------------------------------------------------
__global__ void __launch_bounds__(256)
vox_zero(float* __restrict__ out, int n) {
    int n4 = n >> 2;
    int i = blockIdx.x * blockDim.x + threadIdx.x;
    if (i < n4) {
        reinterpret_cast<float4*>(out)[i] = make_float4(0.f, 0.f, 0.f, 0.f);
    }
    if (i == 0) {
        for (int k = n4 << 2; k < n; ++k) out[k] = 0.f;
    }
}

// ---------------------------------------------------------------------------
// Kernel 2: per-batch params via binary search over the sorted batch array.
// Replicates reference semantics: left/right searchsorted, clip, bump when
// cnt>=2 && t[last]==t[first], denom==0 -> 1.
// ---------------------------------------------------------------------------
__global__ void vox_params(const float* __restrict__ pos,
                           const int* __restrict__ batch,
                           int n, BatchParam* __restrict__ bp) {
    int b = threadIdx.x;
    if (b >= B_DIM) return;

    // lower_bound(batch, b)
    int lo = 0, hi = n;
    while (lo < hi) { int mid = (lo + hi) >> 1; if (batch[mid] < b) lo = mid + 1; else hi = mid; }
    int left = lo;
    // lower_bound(batch, b+1) == searchsorted(..., side='right') for b
    lo = 0; hi = n;
    while (lo < hi) { int mid = (lo + hi) >> 1; if (batch[mid] < b + 1) lo = mid + 1; else hi = mid; }
    int right = lo;

    int first = min(max(left, 0), n - 1);
    int last  = min(max(right - 1, 0), n - 1);

    float tf = pos[3 * first + 2];
    float tl = pos[3 * last  + 2];
    int cnt = right - left;
    float bump = (cnt >= 2 && tl == tf) ? 1.0f : 0.0f;
    tl += bump;
    float denom = tl - tf;
    if (denom == 0.0f) denom = 1.0f;

    BatchParam p;
    p.t0 = tf; p.denom = denom; p.bump = bump; p.last = last;
    bp[b] = p;
}

// ---------------------------------------------------------------------------
// Kernel 3: one point per thread, 8 trilinear corner atomics into the grid.
// Non-temporal input loads keep L2 reserved for the output grid.
// Hardware f32 atomic-add (no return) -> global_atomic_add_f32, STOREcnt path.
// ---------------------------------------------------------------------------
__global__ void __launch_bounds__(256)
vox_scatter(const float* __restrict__ pos,
            const float* __restrict__ feat,
            const int* __restrict__ batch,
            const BatchParam* __restrict__ bp,
            float* __restrict__ out, int n) {
    int i = blockIdx.x * blockDim.x + threadIdx.x;
    if (i >= n) return;

    // Stream-once inputs: non-temporal (NT) loads
    float x = __builtin_nontemporal_load(pos + 3 * i + 0);
    float y = __builtin_nontemporal_load(pos + 3 * i + 1);
    float t = __builtin_nontemporal_load(pos + 3 * i + 2);
    float f = __builtin_nontemporal_load(feat + i);
    int   b = __builtin_nontemporal_load(batch + i);

    BatchParam p = bp[b];  // uniform within nearly every wave (batch sorted)

    float te = t + ((i == p.last) ? p.bump : 0.0f);
    float tn = ((float)(T_DIM - 1) * (te - p.t0)) / p.denom;
    float xf = x * (float)(W_DIM - 1);
    float yf = y * (float)(H_DIM - 1);

    int x0 = (int)xf;
    int y0 = (int)yf;
    int t0 = (int)tn;
    int ch = (f > 0.0f) ? 0 : 1;

    // Per-axis corner weights, computed exactly as the reference: 1 - |corner - coord|
    float wx[2], wy[2], wt[2];
#pragma unroll
    for (int d = 0; d < 2; ++d) {
        wx[d] = 1.0f - fabsf((float)(x0 + d) - xf);
        wy[d] = 1.0f - fabsf((float)(y0 + d) - yf);
        wt[d] = 1.0f - fabsf((float)(t0 + d) - tn);
    }

    const int chBase = ((b * 2 + ch) * H_DIM);  // partial index (per t-slice stride applied below)

#pragma unroll
    for (int dt = 0; dt < 2; ++dt) {
        int tl = t0 + dt;
        if ((unsigned)tl >= (unsigned)T_DIM) continue;
#pragma unroll
        for (int dy = 0; dy < 2; ++dy) {
            int yl = y0 + dy;
            if ((unsigned)yl >= (unsigned)H_DIM) continue;
            float wty = wt[dt] * wy[dy];
#pragma unroll
            for (int dx = 0; dx < 2; ++dx) {
                int xl = x0 + dx;
                if ((unsigned)xl >= (unsigned)W_DIM) continue;
                int idx = ((tl * B_DIM * 2 * H_DIM) + chBase + yl) * W_DIM + xl;
                // hardware global_atomic_add_f32 (no return)
                unsafeAtomicAdd(out + idx, wty * wx[dx]);
            }
        }
    }
}

// ---------------------------------------------------------------------------
extern "C" void kernel_launch(void* const* d_in, const int* in_sizes, int n_in,
                              void* d_out, int out_size, void* d_ws, size_t ws_size,
                              hipStream_t stream) {
    const float* pos   = (const float*)d_in[0];  // (N,3) x,y,t
    const float* feat  = (const float*)d_in[1];  // (N,1)
    const int*   batch = (const int*)d_in[2];    // (N,) sorted
    float*       out   = (float*)d_out;          // T*B*2*H*W floats
    BatchParam*  bp    = (BatchParam*)d_ws;      // 8 * 16 bytes

    int n = in_sizes[2];  // N from batch array

    // 1) zero/prewarm output grid
    int n4 = out_size >> 2;
    int zb = (n4 + 255) / 256;
    vox_zero<<<zb, 256, 0, stream>>>(out, out_size);

    // 2) per-batch params (one wave)
    vox_params<<<1, 32, 0, stream>>>(pos, batch, n, bp);

    // 3) scatter
    int sb = (n + 255) / 256;
    vox_scatter<<<sb, 256, 0, stream>>>(pos, feat, batch, bp, out, n);
}